// GCN_30262339568120
// MI455X (gfx1250) — compile-verified
//
#include <hip/hip_runtime.h>
#include <hip/hip_bf16.h>

typedef __attribute__((ext_vector_type(2))) float v2f;
typedef __attribute__((ext_vector_type(8))) float v8f;

static constexpr int DIM        = 64;
static constexpr int EMB_STRIDE = 256;   // (L+1)*DIM
static constexpr int NUSER      = 100000;
static constexpr float NEG_SLOPE = 0.01f;

// ---------------------------------------------------------------- utilities

__global__ __launch_bounds__(256) void zero_kernel(float* __restrict__ p, int n) {
  int i = blockIdx.x * blockDim.x + threadIdx.x;
  if (i < n) p[i] = 0.0f;
}

// copy features[N,64] into final_emb[:, 0:64] (row stride 256)
__global__ __launch_bounds__(256) void copy_slice0_kernel(const float* __restrict__ feat,
                                                          float* __restrict__ femb,
                                                          int n_elems) {
  int i = blockIdx.x * blockDim.x + threadIdx.x;
  if (i >= n_elems) return;
  int node = i >> 6, d = i & 63;
  femb[(size_t)node * EMB_STRIDE + d] = feat[i];
}

// ---------------------------------------------------------------- SpMM (COO)
// One wave per edge; each lane owns 2 of 64 dims. feats slice is L2-resident
// (38.4MB << 192MB L2) so the gather and the f32 atomics stay on-chip.
__global__ __launch_bounds__(256) void spmm_kernel(const int* __restrict__ rows,
                                                   const int* __restrict__ cols,
                                                   const float* __restrict__ vals,
                                                   const float* __restrict__ feats, // stride EMB_STRIDE
                                                   float* __restrict__ agg,         // [N,64]
                                                   int nedges) {
  unsigned gid = blockIdx.x * blockDim.x + threadIdx.x;
  int e = (int)(gid >> 5);           // wave-uniform
  if (e >= nedges) return;
  int lane = gid & 31;
  int r = rows[e];
  int c = cols[e];
  float v = vals[e];
  int d0 = lane * 2;
  const float* x = feats + (size_t)c * EMB_STRIDE + d0;
  float* dst = agg + (size_t)r * DIM + d0;
  atomicAdd(dst + 0, v * x[0]);
  atomicAdd(dst + 1, v * x[1]);
}

// ---------------------------------------------------------------- dense layer
// One wave computes a 16-node tile:
//   out[16,64] = leaky((agg+f)@W1^T + (agg*f)@W2^T + b1 + b2)
// via V_WMMA_F32_16X16X4_F32 chained over K=64 (16 steps), 4 N-tiles, 2 GEMMs.
__global__ __launch_bounds__(256) void layer_kernel(const float* __restrict__ feats, // stride EMB_STRIDE
                                                    const float* __restrict__ agg,   // [N,64]
                                                    const float* __restrict__ W1,    // [64,64]
                                                    const float* __restrict__ b1,    // [64]
                                                    const float* __restrict__ W2,
                                                    const float* __restrict__ b2,
                                                    float* __restrict__ outp,        // stride EMB_STRIDE
                                                    int n_nodes) {
  int wave = (blockIdx.x * blockDim.x + threadIdx.x) >> 5;
  int lane = threadIdx.x & 31;
  int m0 = wave * 16;
  if (m0 >= n_nodes) return;                 // wave-uniform; EXEC stays all-ones for WMMA

  int ml = lane & 15;                        // A row / B,C column within tile
  int kh = (lane >> 4) * 2;                  // K sub-offset for this half-wave
  int node = m0 + ml;
  const float* xf = feats + (size_t)node * EMB_STRIDE;
  const float* xa = agg   + (size_t)node * DIM;

  v8f acc[4];
  v8f z = {};
  for (int nt = 0; nt < 4; ++nt) acc[nt] = z;

  for (int ks = 0; ks < 16; ++ks) {
    int k = ks * 4 + kh;
    float f0 = xf[k], f1 = xf[k + 1];
    float g0 = xa[k], g1 = xa[k + 1];
    v2f a1; a1.x = g0 + f0; a1.y = g1 + f1;   // (agg + feats) fragment
    v2f a2; a2.x = g0 * f0; a2.y = g1 * f1;   // (agg * feats) fragment
    for (int nt = 0; nt < 4; ++nt) {
      int n = nt * 16 + ml;
      v2f bw1; bw1.x = W1[n * DIM + k]; bw1.y = W1[n * DIM + k + 1];  // B[k][n]=W[n][k]
      v2f bw2; bw2.x = W2[n * DIM + k]; bw2.y = W2[n * DIM + k + 1];
      acc[nt] = __builtin_amdgcn_wmma_f32_16x16x4_f32(false, a1, false, bw1,
                                                      (short)0, acc[nt], false, false);
      acc[nt] = __builtin_amdgcn_wmma_f32_16x16x4_f32(false, a2, false, bw2,
                                                      (short)0, acc[nt], false, false);
    }
  }

  // C/D layout: VGPR j, lane L -> row = j + 8*(L/16), col = L%16 (+16*nt)
  int mbase = m0 + (lane >> 4) * 8;
  for (int nt = 0; nt < 4; ++nt) {
    int n = nt * 16 + ml;
    float bias = b1[n] + b2[n];
    for (int j = 0; j < 8; ++j) {
      float v = acc[nt][j] + bias;
      v = (v > 0.0f) ? v : v * NEG_SLOPE;
      outp[(size_t)(mbase + j) * EMB_STRIDE + n] = v;
    }
  }
}

// ---------------------------------------------------------------- scoring
// One wave per pair: project user & item embeddings through Wt[64,256]+bt,
// dot the results. Tiny (0.27 GFLOP total) -> simple VALU + shuffle reduce.
__global__ __launch_bounds__(256) void score_kernel(const float* __restrict__ femb,
                                                    const int* __restrict__ users,
                                                    const int* __restrict__ items,
                                                    const float* __restrict__ Wt,  // [64,256]
                                                    const float* __restrict__ bt,  // [64]
                                                    float* __restrict__ out, int nB) {
  int wave = (blockIdx.x * blockDim.x + threadIdx.x) >> 5;
  int lane = threadIdx.x & 31;
  if (wave >= nB) return;
  const float* urow = femb + (size_t)users[wave] * EMB_STRIDE;
  const float* irow = femb + (size_t)(NUSER + items[wave]) * EMB_STRIDE;
  float partial = 0.0f;
  for (int nn = 0; nn < 2; ++nn) {
    int n = lane * 2 + nn;
    const float* w = Wt + n * EMB_STRIDE;
    float su = bt[n], si = bt[n];
    for (int k = 0; k < EMB_STRIDE; ++k) {
      float wv = w[k];
      su += urow[k] * wv;
      si += irow[k] * wv;
    }
    partial += su * si;
  }
  for (int off = 16; off > 0; off >>= 1)
    partial += __shfl_down(partial, off);
  if (lane == 0) out[wave] = partial;
}

// ---------------------------------------------------------------- launch

extern "C" void kernel_launch(void* const* d_in, const int* in_sizes, int n_in,
                              void* d_out, int out_size, void* d_ws, size_t ws_size,
                              hipStream_t stream) {
  const float* features = (const float*)d_in[0];
  const int*   rows     = (const int*)  d_in[1];
  const int*   cols     = (const int*)  d_in[2];
  const float* vals     = (const float*)d_in[3];
  const int*   users    = (const int*)  d_in[4];
  const int*   items    = (const int*)  d_in[5];
  const float* W1       = (const float*)d_in[6];   // [L,64,64]
  const float* b1       = (const float*)d_in[7];   // [L,64]
  const float* W2       = (const float*)d_in[8];
  const float* b2       = (const float*)d_in[9];
  const float* Wt       = (const float*)d_in[10];  // [64,256]
  const float* bt       = (const float*)d_in[11];  // [64]

  const int N = in_sizes[0] / DIM;        // 150000
  const int E = in_sizes[1];              // 2400000
  const int B = in_sizes[4];              // 4096
  const int L = in_sizes[7] / DIM;        // 3

  // workspace: agg [N,64] then final_emb [N,256]
  float* agg  = (float*)d_ws;
  float* femb = (float*)((char*)d_ws + (size_t)N * DIM * sizeof(float));
  float* outf = (float*)d_out;

  // emb0 = features
  {
    int n_elems = N * DIM;
    copy_slice0_kernel<<<(n_elems + 255) / 256, 256, 0, stream>>>(features, femb, n_elems);
  }

  for (int l = 0; l < L; ++l) {
    const float* fcur = femb + (size_t)l * DIM;        // column slice l, stride 256
    float*       fnext = femb + (size_t)(l + 1) * DIM; // column slice l+1
    {
      int n_elems = N * DIM;
      zero_kernel<<<(n_elems + 255) / 256, 256, 0, stream>>>(agg, n_elems);
    }
    {
      long long threads = (long long)E * 32;
      int grid = (int)((threads + 255) / 256);
      spmm_kernel<<<grid, 256, 0, stream>>>(rows, cols, vals, fcur, agg, E);
    }
    {
      int tiles = (N + 15) / 16;          // 9375
      int grid = (tiles + 7) / 8;         // 8 waves / block
      layer_kernel<<<grid, 256, 0, stream>>>(fcur, agg,
                                             W1 + (size_t)l * DIM * DIM, b1 + (size_t)l * DIM,
                                             W2 + (size_t)l * DIM * DIM, b2 + (size_t)l * DIM,
                                             fnext, N);
    }
  }

  {
    int grid = (B + 7) / 8;               // 8 waves / block, one wave per pair
    score_kernel<<<grid, 256, 0, stream>>>(femb, users, items, Wt, bt, outf, B);
  }
}